// Dynamic_MultiTeacher2_52699248721944
// MI455X (gfx1250) — compile-verified
//
#include <hip/hip_runtime.h>
#include <math.h>

// ---------------------------------------------------------------------------
// Multi-teacher KD loss, single-pass streaming kernel for gfx1250.
// 3 fp32 matrices (2048x32000) streamed exactly once (~768MB -> ~34us at
// 23.3 TB/s HBM). T=20 softmax sums accumulated with NO max-shift (exp(v/20)
// cannot overflow fp32 for |v| < ~1500), so the hot loop is branch-free
// except the T=1 CE logsumexp. Async global->LDS double-buffer pipeline
// (ASYNCcnt-tracked, lane-private 16B slices -> zero barriers).
// ---------------------------------------------------------------------------

#define WG 256
#define TILE4 WG          // float4 slots per tile per input
#define TILE_F (WG * 4)   // floats per tile (1024)

#define L2E_F 1.4426950408889634f   // log2(e)
#define K20_F 0.07213475204444817f  // log2(e)/20
#define K10_F 0.03606737602222408f  // log2(e)/40  (mimic tracked as v1+v2)

// native base-2 exp (v_exp_f32)
#if defined(__has_builtin)
#if __has_builtin(__builtin_amdgcn_exp2f)
#define FEXP2(x) __builtin_amdgcn_exp2f(x)
#endif
#endif
#ifndef FEXP2
#define FEXP2(x) exp2f(x)
#endif

#if defined(__has_builtin)
#if __has_builtin(__builtin_amdgcn_global_load_async_to_lds_b128)
#define HAVE_ASYNC 1
#endif
#endif

#if defined(HAVE_ASYNC)
#if __has_builtin(__builtin_amdgcn_s_wait_asynccnt)
#define WAIT_ASYNC(n) __builtin_amdgcn_s_wait_asynccnt(n)
#else
#define WAIT_ASYNC(n) asm volatile("s_wait_asynccnt %0" ::"i"(n) : "memory")
#endif

typedef int v4i __attribute__((ext_vector_type(4)));
typedef __attribute__((address_space(1))) v4i* as1_v4i;
typedef __attribute__((address_space(3))) v4i* as3_v4i;

__device__ __forceinline__ void async_ld16(const float* g, float4* lds_slot) {
  // one 16B async copy, lane-private LDS destination -> no barriers needed,
  // ordering enforced purely with ASYNCcnt.
  __builtin_amdgcn_global_load_async_to_lds_b128(
      (as1_v4i)(v4i*)(void*)const_cast<float*>(g),
      (as3_v4i)(v4i*)(void*)lds_slot, 0, 0);
}
#endif

struct Acc {
  float ms, z1;          // student T=1: online-max logsumexp (overflow-safe)
  float z20;             // student T=20: raw  sum exp2(s*K20)
  float m1, t21, zA, swA;  // teacher o1: top1/top2, sum e, sum e*s
  float m2, t22, zB, swB;  // teacher o2
  float m3, t23, zC, swC;  // mimic, top1/top2 tracked in (v1+v2) units
};

__device__ __forceinline__ void acc_init(Acc& a) {
  const float ninf = -__builtin_inff();
  a.ms = ninf; a.z1 = 0.f; a.z20 = 0.f;
  a.m1 = ninf; a.t21 = ninf; a.zA = 0.f; a.swA = 0.f;
  a.m2 = ninf; a.t22 = ninf; a.zB = 0.f; a.swB = 0.f;
  a.m3 = ninf; a.t23 = ninf; a.zC = 0.f; a.swC = 0.f;
}

__device__ __forceinline__ void acc_update(Acc& a, float v1, float v2, float vs) {
  // ---- student CE stream (T=1): online max, the only branch ----
  if (vs > a.ms) {
    a.z1 *= FEXP2((a.ms - vs) * L2E_F);  // -inf first time: 0*0
    a.ms = vs;
  }
  a.z1 += FEXP2((vs - a.ms) * L2E_F);
  // ---- student T=20 logsumexp: raw (no shift needed) ----
  a.z20 += FEXP2(vs * K20_F);
  // ---- teacher 1 ----
  float e1 = FEXP2(v1 * K20_F);
  a.zA += e1;
  a.swA = fmaf(e1, vs, a.swA);
  a.t21 = fmaxf(a.t21, fminf(a.m1, v1));  // branchless top-2
  a.m1 = fmaxf(a.m1, v1);
  // ---- teacher 2 ----
  float e2 = FEXP2(v2 * K20_F);
  a.zB += e2;
  a.swB = fmaf(e2, vs, a.swB);
  a.t22 = fmaxf(a.t22, fminf(a.m2, v2));
  a.m2 = fmaxf(a.m2, v2);
  // ---- mimic: tracked in t=(v1+v2) units; exp2(t*K10)==exp(mimic/20) ----
  float t = v1 + v2;
  float em = FEXP2(t * K10_F);
  a.zC += em;
  a.swC = fmaf(em, vs, a.swC);
  a.t23 = fmaxf(a.t23, fminf(a.m3, t));
  a.m3 = fmaxf(a.m3, t);
}

// merge partner's stats: only z1 needs max-rescaling, everything else is
// plain add / top-2 pair merge.
__device__ __forceinline__ void merge_z1(float& ms, float& z1, float bms, float bz1) {
  float nm = fmaxf(ms, bms);
  z1 = z1 * FEXP2((ms - nm) * L2E_F) + bz1 * FEXP2((bms - nm) * L2E_F);
  ms = nm;
}
__device__ __forceinline__ void merge_top2(float& m, float& t2, float bm, float bt2) {
  float nt2 = fmaxf(fminf(m, bm), (m >= bm) ? t2 : bt2);
  m = fmaxf(m, bm);
  t2 = nt2;
}
__device__ __forceinline__ void merge_all(Acc& a, const Acc& b) {
  merge_z1(a.ms, a.z1, b.ms, b.z1);
  a.z20 += b.z20;
  merge_top2(a.m1, a.t21, b.m1, b.t21);
  a.zA += b.zA; a.swA += b.swA;
  merge_top2(a.m2, a.t22, b.m2, b.t22);
  a.zB += b.zB; a.swB += b.swB;
  merge_top2(a.m3, a.t23, b.m3, b.t23);
  a.zC += b.zC; a.swC += b.swC;
}

__global__ void __launch_bounds__(WG) kd_row_kernel(
    const float* __restrict__ o1, const float* __restrict__ o2,
    const float* __restrict__ os, const int* __restrict__ tgts,
    float* __restrict__ wsA, float* __restrict__ wsB, float* __restrict__ wsM,
    int C) {
  const int row = blockIdx.x;
  const int tid = threadIdx.x;
  const float* r1 = o1 + (size_t)row * C;
  const float* r2 = o2 + (size_t)row * C;
  const float* rs = os + (size_t)row * C;

  Acc a;
  acc_init(a);

#if defined(HAVE_ASYNC)
  // double-buffered async global->LDS pipeline (lane-private 16B slices)
  __shared__ float4 tile[2 * 3 * TILE4];  // 24 KB
  const int nT = C / TILE_F;
  if (nT > 0) {
    {
      const size_t o = (size_t)tid * 4;
      async_ld16(r1 + o, &tile[(0 * 3 + 0) * TILE4 + tid]);
      async_ld16(r2 + o, &tile[(0 * 3 + 1) * TILE4 + tid]);
      async_ld16(rs + o, &tile[(0 * 3 + 2) * TILE4 + tid]);
    }
    if (nT > 1) {
      const size_t o = (size_t)TILE_F + (size_t)tid * 4;
      async_ld16(r1 + o, &tile[(1 * 3 + 0) * TILE4 + tid]);
      async_ld16(r2 + o, &tile[(1 * 3 + 1) * TILE4 + tid]);
      async_ld16(rs + o, &tile[(1 * 3 + 2) * TILE4 + tid]);
    }
    for (int t = 0; t < nT; ++t) {
      if (t + 1 < nT) {
        WAIT_ASYNC(3);  // 6 in flight, in-order: first 3 (this buffer) done
      } else {
        WAIT_ASYNC(0);
      }
      const int b = t & 1;
      float4 a1 = tile[(b * 3 + 0) * TILE4 + tid];
      float4 a2 = tile[(b * 3 + 1) * TILE4 + tid];
      float4 av = tile[(b * 3 + 2) * TILE4 + tid];
      acc_update(a, a1.x, a2.x, av.x);
      acc_update(a, a1.y, a2.y, av.y);
      acc_update(a, a1.z, a2.z, av.z);
      acc_update(a, a1.w, a2.w, av.w);
      const int nx = t + 2;
      if (nx < nT) {
        const size_t o = (size_t)nx * TILE_F + (size_t)tid * 4;
        async_ld16(r1 + o, &tile[(b * 3 + 0) * TILE4 + tid]);
        async_ld16(r2 + o, &tile[(b * 3 + 1) * TILE4 + tid]);
        async_ld16(rs + o, &tile[(b * 3 + 2) * TILE4 + tid]);
      }
    }
  }
  // tail (C % 1024 floats), direct loads
  {
    const int base = nT * TILE_F + tid * 4;
    if (base < C) {
      float4 a1 = *(const float4*)(r1 + base);
      float4 a2 = *(const float4*)(r2 + base);
      float4 av = *(const float4*)(rs + base);
      acc_update(a, a1.x, a2.x, av.x);
      acc_update(a, a1.y, a2.y, av.y);
      acc_update(a, a1.z, a2.z, av.z);
      acc_update(a, a1.w, a2.w, av.w);
    }
  }
#else
  // fallback: plain float4 streaming
  const int n4 = C >> 2;
  for (int j = tid; j < n4; j += WG) {
    float4 a1 = ((const float4*)r1)[j];
    float4 a2 = ((const float4*)r2)[j];
    float4 av = ((const float4*)rs)[j];
    acc_update(a, a1.x, a2.x, av.x);
    acc_update(a, a1.y, a2.y, av.y);
    acc_update(a, a1.z, a2.z, av.z);
    acc_update(a, a1.w, a2.w, av.w);
  }
#endif

  // ---------------- wave32 butterfly reduction ----------------
#pragma unroll
  for (int off = 16; off; off >>= 1) {
    Acc b;
    b.ms = __shfl_xor(a.ms, off, 32);
    b.z1 = __shfl_xor(a.z1, off, 32);
    b.z20 = __shfl_xor(a.z20, off, 32);
    b.m1 = __shfl_xor(a.m1, off, 32);
    b.t21 = __shfl_xor(a.t21, off, 32);
    b.zA = __shfl_xor(a.zA, off, 32);
    b.swA = __shfl_xor(a.swA, off, 32);
    b.m2 = __shfl_xor(a.m2, off, 32);
    b.t22 = __shfl_xor(a.t22, off, 32);
    b.zB = __shfl_xor(a.zB, off, 32);
    b.swB = __shfl_xor(a.swB, off, 32);
    b.m3 = __shfl_xor(a.m3, off, 32);
    b.t23 = __shfl_xor(a.t23, off, 32);
    b.zC = __shfl_xor(a.zC, off, 32);
    b.swC = __shfl_xor(a.swC, off, 32);
    merge_all(a, b);
  }

  // ---------------- cross-wave via LDS (8 waves) ----------------
  __shared__ Acc red[8];
  const int wave = tid >> 5;
  if ((tid & 31) == 0) red[wave] = a;
  __syncthreads();

  if (tid == 0) {
    for (int w = 1; w < 8; ++w) merge_all(a, red[w]);

    // target logits fetched directly (cheaper than scanning the stream)
    const int tgt = tgts[row];
    const float st = rs[tgt];
    const float a1 = r1[tgt];
    const float a2 = r2[tgt];
    const float tsum = a1 + a2;         // same expr as streamed mimic basis
    const float am = 0.5f * tsum;

    const float CE = -(st - a.ms - logf(a.z1));
    const float lZ = logf(a.z20);       // logsumexp(s/20), reference 0
    const float KD0 = 400.f * (lZ - (a.swA / a.zA) * 0.05f);
    const float KD1 = 400.f * (lZ - (a.swB / a.zB) * 0.05f);
    const float KD2 = 400.f * (lZ - (a.swC / a.zC) * 0.05f);

    // margins: top1-top2 iff target holds the row max (value equality,
    // mimic compared in exact 2x units)
    const float d0 = (a1 == a.m1) ? (a.m1 - a.t21) : 0.f;
    const float d1 = (a2 == a.m2) ? (a.m2 - a.t22) : 0.f;
    const float d2 = (tsum == a.m3) ? 0.5f * (a.m3 - a.t23) : 0.f;

    // threshold = softmax(d / 2)
    const float h0 = d0 * 0.5f, h1 = d1 * 0.5f, h2 = d2 * 0.5f;
    const float hm = fmaxf(h0, fmaxf(h1, h2));
    const float e0 = FEXP2((h0 - hm) * L2E_F);
    const float e1 = FEXP2((h1 - hm) * L2E_F);
    const float e2 = FEXP2((h2 - hm) * L2E_F);
    const float inv = 1.f / (e0 + e1 + e2);

    wsA[row] = CE;
    wsB[row] = 0.8f * (e0 * inv * a1 * (KD0 - CE) +
                       e1 * inv * a2 * (KD1 - CE) +
                       e2 * inv * am * (KD2 - CE));
    wsM[row] = fmaxf(a.m1, a.m2);  // candidate for global max_preds
  }
}

// loss_b = CE_b + B_b / max_preds ; output = mean over rows
__global__ void __launch_bounds__(256) kd_finalize_kernel(
    const float* __restrict__ wsA, const float* __restrict__ wsB,
    const float* __restrict__ wsM, float* __restrict__ out, int rows) {
  __shared__ float sm[256];
  const int tid = threadIdx.x;
  float mx = -__builtin_inff();
  for (int i = tid; i < rows; i += 256) mx = fmaxf(mx, wsM[i]);
  sm[tid] = mx;
  __syncthreads();
  for (int s = 128; s; s >>= 1) {
    if (tid < s) sm[tid] = fmaxf(sm[tid], sm[tid + s]);
    __syncthreads();
  }
  const float inv_maxp = 1.f / sm[0];
  __syncthreads();
  float acc = 0.f;
  for (int i = tid; i < rows; i += 256) acc += wsA[i] + wsB[i] * inv_maxp;
  sm[tid] = acc;
  __syncthreads();
  for (int s = 128; s; s >>= 1) {
    if (tid < s) sm[tid] += sm[tid + s];
    __syncthreads();
  }
  if (tid == 0) out[0] = sm[0] / (float)rows;
}

extern "C" void kernel_launch(void* const* d_in, const int* in_sizes, int n_in,
                              void* d_out, int out_size, void* d_ws, size_t ws_size,
                              hipStream_t stream) {
  (void)n_in;
  (void)out_size;
  (void)ws_size;
  const float* o1 = (const float*)d_in[0];
  const float* o2 = (const float*)d_in[1];
  const float* os = (const float*)d_in[2];
  const int* tg = (const int*)d_in[3];
  const int rows = in_sizes[3];
  const int C = in_sizes[0] / rows;

  float* wsA = (float*)d_ws;
  float* wsB = wsA + rows;
  float* wsM = wsB + rows;

  kd_row_kernel<<<rows, WG, 0, stream>>>(o1, o2, os, tg, wsA, wsB, wsM, C);
  kd_finalize_kernel<<<1, 256, 0, stream>>>(wsA, wsB, wsM, (float*)d_out, rows);
}